// GCN1_62173946577682
// MI455X (gfx1250) — compile-verified
//
#include <hip/hip_runtime.h>
#include <hip/hip_bf16.h>

typedef __attribute__((ext_vector_type(2))) float v2f;
typedef __attribute__((ext_vector_type(8))) float v8f;

#define N_NODES 100000
#define N_EDGES 800000
#define F_IN    128
#define HID     512
#define N_CLS   40
#define BN_EPS  1e-5f
#define ROWS_PER_CHUNK 256
#define NCH ((N_NODES + ROWS_PER_CHUNK - 1) / ROWS_PER_CHUNK)

// ---------------------------------------------------------------------------
// WMMA fp32 GEMM: C[M x Nc] = A[M x K] @ B[K x Nc]
// block = 256 threads (8 waves). Block tile: 16 rows x 256 cols.
// Each wave owns a 16x32 slice (2 accumulators) -> A-frag reused across 2 WMMAs.
// K staged through LDS in chunks of 32; per chunk all A/B fragments are
// preloaded to registers, then 16 V_WMMA_F32_16X16X4_F32 issue back-to-back.
// Requires: M % 16 == 0, K % 32 == 0, Nc % 4 == 0 (B zero-padded, stores guarded).
// ---------------------------------------------------------------------------
__global__ __launch_bounds__(256) void wmma_gemm_f32(
    const float* __restrict__ A, const float* __restrict__ B,
    float* __restrict__ C, int M, int K, int Nc)
{
    __shared__ __align__(16) float As[16][36];    // 16 x 32, stride 36 (aligned, conflict-free)
    __shared__ __align__(16) float Bs[32][260];   // 32 x 256, stride 260

    const int tid  = threadIdx.x;
    const int wave = tid >> 5;
    const int lane = tid & 31;
    const int half = lane >> 4;          // 0 or 1
    const int l16  = lane & 15;
    const int cw   = wave * 32;          // wave's col offset inside block tile

    const int row0     = blockIdx.y * 16;
    const int colBlock = blockIdx.x * 256;

    v8f acc0 = {0.f, 0.f, 0.f, 0.f, 0.f, 0.f, 0.f, 0.f};
    v8f acc1 = {0.f, 0.f, 0.f, 0.f, 0.f, 0.f, 0.f, 0.f};

    for (int k0 = 0; k0 < K; k0 += 32) {
        // ---- A tile: 16x32 = 128 float4, threads 0..127 load one each ----
        if (tid < 128) {
            int ar  = tid >> 3;          // row 0..15
            int ac4 = tid & 7;           // float4 col 0..7
            float4 v = *reinterpret_cast<const float4*>(
                A + (size_t)(row0 + ar) * K + (k0 + ac4 * 4));
            *reinterpret_cast<float4*>(&As[ar][ac4 * 4]) = v;
        }
        // ---- B tile: 32x256 = 2048 float4, 8 per thread (zero-pad past Nc) ----
        #pragma unroll
        for (int i = 0; i < 8; ++i) {
            int e   = tid + i * 256;     // 0..2047
            int br  = e >> 6;            // row 0..31
            int bc4 = e & 63;            // float4 col 0..63
            int gc  = colBlock + bc4 * 4;
            float4 v = make_float4(0.f, 0.f, 0.f, 0.f);
            if (gc < Nc)
                v = *reinterpret_cast<const float4*>(B + (size_t)(k0 + br) * Nc + gc);
            *reinterpret_cast<float4*>(&Bs[br][bc4 * 4]) = v;
        }
        __syncthreads();

        // ---- preload all fragments for this K-chunk ----
        v2f af[8], bf0[8], bf1[8];
        #pragma unroll
        for (int q = 0; q < 8; ++q) {
            // A frag (16x4 MxK): lanes0-15 M=0..15, half selects K pair {2h, 2h+1}
            af[q].x  = As[l16][4 * q + 2 * half + 0];
            af[q].y  = As[l16][4 * q + 2 * half + 1];
            // B frag (4x16 KxN): row striped across lanes in VGPR, K = 2h + v
            bf0[q].x = Bs[4 * q + 2 * half + 0][cw + l16];
            bf0[q].y = Bs[4 * q + 2 * half + 1][cw + l16];
            bf1[q].x = Bs[4 * q + 2 * half + 0][cw + 16 + l16];
            bf1[q].y = Bs[4 * q + 2 * half + 1][cw + 16 + l16];
        }
        // ---- 16 WMMAs, two independent accumulator chains ----
        #pragma unroll
        for (int q = 0; q < 8; ++q) {
            acc0 = __builtin_amdgcn_wmma_f32_16x16x4_f32(
                false, af[q], false, bf0[q], (short)0, acc0, false, false);
            acc1 = __builtin_amdgcn_wmma_f32_16x16x4_f32(
                false, af[q], false, bf1[q], (short)0, acc1, false, false);
        }
        __syncthreads();
    }

    // C/D layout: VGPR r -> row = r + 8*half, col = l16
    const int gcol0 = colBlock + cw + l16;
    const int gcol1 = gcol0 + 16;
    if (gcol0 < Nc) {
        #pragma unroll
        for (int r = 0; r < 8; ++r)
            C[(size_t)(row0 + r + 8 * half) * Nc + gcol0] = acc0[r];
    }
    if (gcol1 < Nc) {
        #pragma unroll
        for (int r = 0; r < 8; ++r)
            C[(size_t)(row0 + r + 8 * half) * Nc + gcol1] = acc1[r];
    }
}

// ---------------------------------------------------------------------------
// Degree / normalization
// ---------------------------------------------------------------------------
__global__ __launch_bounds__(256) void zero_f32(float* p, int n) {
    int i = blockIdx.x * blockDim.x + threadIdx.x;
    if (i < n) p[i] = 0.f;
}

__global__ __launch_bounds__(256) void deg_count(const int* __restrict__ dst,
                                                 float* deg, int E) {
    int e = blockIdx.x * blockDim.x + threadIdx.x;
    if (e < E) atomicAdd(deg + dst[e], 1.0f);
}

__global__ __launch_bounds__(256) void make_dinv(float* deg, int n) {
    int i = blockIdx.x * blockDim.x + threadIdx.x;
    if (i < n) deg[i] = rsqrtf(deg[i] + 1.0f);   // deg^-1/2 incl. self-loop
}

// ---------------------------------------------------------------------------
// agg init: agg = h * dinv^2 + bias   (self-loop term + bias)
// ---------------------------------------------------------------------------
__global__ __launch_bounds__(256) void self_bias(
    const float* __restrict__ h, const float* __restrict__ dinv,
    const float* __restrict__ bias, float* __restrict__ agg, int Nrows, int F)
{
    size_t i = (size_t)blockIdx.x * blockDim.x + threadIdx.x;
    size_t total = (size_t)Nrows * F;
    if (i >= total) return;
    int row = (int)(i / F);
    int col = (int)(i % F);
    float d = dinv[row];
    agg[i] = h[i] * d * d + bias[col];
}

// ---------------------------------------------------------------------------
// Edge scatter: one wave per edge; float4 gathers, per-float f32 atomics.
// ---------------------------------------------------------------------------
__global__ __launch_bounds__(256) void edge_agg(
    const float* __restrict__ h, const int* __restrict__ src,
    const int* __restrict__ dst, const float* __restrict__ dinv,
    float* __restrict__ agg, int E, int F)
{
    int e = blockIdx.x * (blockDim.x >> 5) + (threadIdx.x >> 5);
    if (e >= E) return;
    int lane = threadIdx.x & 31;
    int s = src[e], d = dst[e];
    float nrm = dinv[s] * dinv[d];
    const float* hs = h + (size_t)s * F;
    float* ad = agg + (size_t)d * F;
    for (int f = lane * 4; f < F; f += 128) {
        float4 v = *reinterpret_cast<const float4*>(hs + f);
        atomicAdd(ad + f + 0, v.x * nrm);
        atomicAdd(ad + f + 1, v.y * nrm);
        atomicAdd(ad + f + 2, v.z * nrm);
        atomicAdd(ad + f + 3, v.w * nrm);
    }
}

// ---------------------------------------------------------------------------
// BatchNorm (training mode, biased var): partials -> finalize -> affine apply
// ---------------------------------------------------------------------------
__global__ __launch_bounds__(256) void bn_partial(
    const float* __restrict__ x, float* __restrict__ psum,
    float* __restrict__ psumsq, int Nrows, int F)
{
    int chunk = blockIdx.x;
    int r0 = chunk * ROWS_PER_CHUNK;
    int r1 = min(r0 + ROWS_PER_CHUNK, Nrows);
    for (int col = threadIdx.x; col < F; col += blockDim.x) {
        float s = 0.f, sq = 0.f;
        for (int r = r0; r < r1; ++r) {
            float v = x[(size_t)r * F + col];
            s += v; sq += v * v;
        }
        psum  [(size_t)chunk * F + col] = s;
        psumsq[(size_t)chunk * F + col] = sq;
    }
}

__global__ __launch_bounds__(256) void bn_finalize(
    const float* __restrict__ psum, const float* __restrict__ psumsq,
    const float* __restrict__ gamma, const float* __restrict__ beta,
    float* coefA, float* coefC, int nchunks, int F, float invN)
{
    int col = blockIdx.x * blockDim.x + threadIdx.x;
    if (col >= F) return;
    float s = 0.f, sq = 0.f;
    for (int c = 0; c < nchunks; ++c) {
        s  += psum  [(size_t)c * F + col];
        sq += psumsq[(size_t)c * F + col];
    }
    float mu  = s * invN;
    float var = sq * invN - mu * mu;
    float a = gamma[col] * rsqrtf(var + BN_EPS);
    coefA[col] = a;
    coefC[col] = beta[col] - a * mu;
}

__global__ __launch_bounds__(256) void bn_apply(
    const float* __restrict__ x, const float* __restrict__ coefA,
    const float* __restrict__ coefC, float* __restrict__ y,
    size_t total, int F, int relu)
{
    size_t i = (size_t)blockIdx.x * blockDim.x + threadIdx.x;
    if (i >= total) return;
    int col = (int)(i % F);
    float v = coefA[col] * x[i] + coefC[col];
    if (relu) v = fmaxf(v, 0.f);
    y[i] = v;
}

// ---------------------------------------------------------------------------
extern "C" void kernel_launch(void* const* d_in, const int* in_sizes, int n_in,
                              void* d_out, int out_size, void* d_ws, size_t ws_size,
                              hipStream_t stream) {
    (void)in_sizes; (void)n_in; (void)out_size; (void)ws_size;

    const float* x   = (const float*)d_in[0];
    const int*   ei  = (const int*)d_in[1];
    const float* W1  = (const float*)d_in[2];
    const float* b1  = (const float*)d_in[3];
    const float* g1  = (const float*)d_in[4];
    const float* be1 = (const float*)d_in[5];
    const float* W2  = (const float*)d_in[6];
    const float* b2  = (const float*)d_in[7];
    const float* g2  = (const float*)d_in[8];
    const float* be2 = (const float*)d_in[9];
    const float* W3  = (const float*)d_in[10];
    const float* b3  = (const float*)d_in[11];
    const float* g3  = (const float*)d_in[12];
    const float* be3 = (const float*)d_in[13];

    const int* srcIdx = ei;
    const int* dstIdx = ei + N_EDGES;

    float* out = (float*)d_out;                       // [N, 40]
    float* x6  = out + (size_t)N_NODES * N_CLS;       // [N, 512] second output

    // workspace layout
    float* buf0   = (float*)d_ws;                     // N*512
    float* buf1   = buf0 + (size_t)N_NODES * HID;     // N*512
    float* dinv   = buf1 + (size_t)N_NODES * HID;     // N
    float* psum   = dinv + N_NODES;                   // NCH*512
    float* psumsq = psum + (size_t)NCH * HID;         // NCH*512
    float* coefA  = psumsq + (size_t)NCH * HID;       // 512
    float* coefC  = coefA + HID;                      // 512

    const int TB = 256;
    const int nodeBlocks  = (N_NODES + TB - 1) / TB;
    const int edgeBlocksT = (N_EDGES + TB - 1) / TB;
    const int edgeBlocksW = (N_EDGES + 7) / 8;        // 8 waves per block
    const size_t totH = (size_t)N_NODES * HID;
    const size_t totC = (size_t)N_NODES * N_CLS;
    const int gridH = (int)((totH + TB - 1) / TB);
    const int gridC = (int)((totC + TB - 1) / TB);
    const float invN = 1.0f / (float)N_NODES;

    // degrees -> dinv
    zero_f32 <<<nodeBlocks, TB, 0, stream>>>(dinv, N_NODES);
    deg_count<<<edgeBlocksT, TB, 0, stream>>>(dstIdx, dinv, N_EDGES);
    make_dinv<<<nodeBlocks, TB, 0, stream>>>(dinv, N_NODES);

    // ---- layer 1: h1 = x @ W1 ; agg1 ; BN+ReLU -> buf0 (x3) ----
    wmma_gemm_f32<<<dim3((HID + 255) / 256, N_NODES / 16), TB, 0, stream>>>(
        x, W1, buf0, N_NODES, F_IN, HID);
    self_bias<<<gridH, TB, 0, stream>>>(buf0, dinv, b1, buf1, N_NODES, HID);
    edge_agg <<<edgeBlocksW, TB, 0, stream>>>(buf0, srcIdx, dstIdx, dinv, buf1, N_EDGES, HID);
    bn_partial <<<NCH, TB, 0, stream>>>(buf1, psum, psumsq, N_NODES, HID);
    bn_finalize<<<(HID + TB - 1) / TB, TB, 0, stream>>>(psum, psumsq, g1, be1,
                                                        coefA, coefC, NCH, HID, invN);
    bn_apply<<<gridH, TB, 0, stream>>>(buf1, coefA, coefC, buf0, totH, HID, 1);

    // ---- layer 2: h2 = x3 @ W2 ; agg2 ; BN+ReLU -> x6 (in d_out) ----
    wmma_gemm_f32<<<dim3((HID + 255) / 256, N_NODES / 16), TB, 0, stream>>>(
        buf0, W2, buf1, N_NODES, HID, HID);
    self_bias<<<gridH, TB, 0, stream>>>(buf1, dinv, b2, buf0, N_NODES, HID);
    edge_agg <<<edgeBlocksW, TB, 0, stream>>>(buf1, srcIdx, dstIdx, dinv, buf0, N_EDGES, HID);
    bn_partial <<<NCH, TB, 0, stream>>>(buf0, psum, psumsq, N_NODES, HID);
    bn_finalize<<<(HID + TB - 1) / TB, TB, 0, stream>>>(psum, psumsq, g2, be2,
                                                        coefA, coefC, NCH, HID, invN);
    bn_apply<<<gridH, TB, 0, stream>>>(buf0, coefA, coefC, x6, totH, HID, 1);

    // ---- layer 3: h3 = x6 @ W3 ; agg3 ; BN -> out ----
    wmma_gemm_f32<<<dim3(1, N_NODES / 16), TB, 0, stream>>>(
        x6, W3, buf1, N_NODES, HID, N_CLS);
    self_bias<<<gridC, TB, 0, stream>>>(buf1, dinv, b3, buf0, N_NODES, N_CLS);
    edge_agg <<<edgeBlocksW, TB, 0, stream>>>(buf1, srcIdx, dstIdx, dinv, buf0, N_EDGES, N_CLS);
    bn_partial <<<NCH, TB, 0, stream>>>(buf0, psum, psumsq, N_NODES, N_CLS);
    bn_finalize<<<1, TB, 0, stream>>>(psum, psumsq, g3, be3,
                                      coefA, coefC, NCH, N_CLS, invN);
    bn_apply<<<gridC, TB, 0, stream>>>(buf0, coefA, coefC, out, totC, N_CLS, 0);
}